// ManifoldInterpolator_65687229825413
// MI455X (gfx1250) — compile-verified
//
#include <hip/hip_runtime.h>
#include <hip/hip_bf16.h>
#include <math.h>

typedef float v2f __attribute__((ext_vector_type(2)));
typedef float v8f __attribute__((ext_vector_type(8)));
typedef int   v4i_vs __attribute__((vector_size(16)));   // matches builtin param type

#define EPSF 1e-6f
#define ROWS 128
#define DM   2048
#define LAT  32
#define LPAD 36   // row stride in dwords: 16B aligned (144B) and conflict-free banks

#define AS1 __attribute__((address_space(1)))
#define AS3 __attribute__((address_space(3)))

// ---- async global->LDS staging (CDNA5), with compile-safe fallback ---------
#if __has_builtin(__builtin_amdgcn_global_load_async_to_lds_b128)
#  define ASYNC_B128(g, l)                                             \
     __builtin_amdgcn_global_load_async_to_lds_b128(                   \
         (AS1 v4i_vs*)(g), (AS3 v4i_vs*)(l), 0, 0)
#  define HAVE_ASYNC_LDS 1
#endif

#if defined(HAVE_ASYNC_LDS)
#  if __has_builtin(__builtin_amdgcn_s_wait_asynccnt)
#    define WAIT_ASYNC0() __builtin_amdgcn_s_wait_asynccnt(0)
#  else
#    define WAIT_ASYNC0() asm volatile("s_wait_asynccnt 0x0" ::: "memory")
#  endif
#endif

__device__ __forceinline__ v8f wmma_f32_k4(v2f a, v2f b, v8f c) {
  // D = A(16x4,f32) * B(4x16,f32) + C(16x16,f32)
  return __builtin_amdgcn_wmma_f32_16x16x4_f32(false, a, false, b, (short)0, c,
                                               false, false);
}

__device__ __forceinline__ float gelu_exact(float x) {
  return 0.5f * x * (1.0f + erff(x * 0.70710678118654752440f));
}

__global__ __launch_bounds__(256) void manifold_fused(
    const float* __restrict__ z0, const float* __restrict__ z1,
    const float* __restrict__ tp,
    const float* __restrict__ We1, const float* __restrict__ be1,
    const float* __restrict__ We2, const float* __restrict__ be2,
    const float* __restrict__ Wd1, const float* __restrict__ bd1,
    const float* __restrict__ Wd2, const float* __restrict__ bd2,
    float* __restrict__ out)
{
  __shared__ float lds[2][2][ROWS][LPAD];   // [buf][z][row][col] = 72 KB

  const int tid   = threadIdx.x;
  const int wave  = tid >> 5;
  const int lane  = tid & 31;
  const int l16   = lane & 15;
  const int hi    = lane >> 4;     // 0: K even pair, 1: K odd pair (A/B frag layout)
  const int ko2   = hi * 2;
  const int blockRow = blockIdx.x * ROWS;
  const int myRow = wave * 16;

  const v8f zero8 = {0.f,0.f,0.f,0.f,0.f,0.f,0.f,0.f};
  v8f acc00 = zero8, acc01 = zero8, acc10 = zero8, acc11 = zero8;

  // One 16x32 K-chunk of WMMAs against lds[buf]
  auto encode_step = [&](int buf, int k0) {
    #pragma unroll
    for (int kk = 0; kk < 32; kk += 4) {
      int ko = kk + ko2;
      v2f a0 = *(const v2f*)&lds[buf][0][myRow + l16][ko];
      v2f a1 = *(const v2f*)&lds[buf][1][myRow + l16][ko];
      int krow = k0 + ko;
      v2f b0, b1;
      b0.x = We1[krow * LAT + l16];
      b0.y = We1[(krow + 1) * LAT + l16];
      b1.x = We1[krow * LAT + 16 + l16];
      b1.y = We1[(krow + 1) * LAT + 16 + l16];
      acc00 = wmma_f32_k4(a0, b0, acc00);
      acc01 = wmma_f32_k4(a0, b1, acc01);
      acc10 = wmma_f32_k4(a1, b0, acc10);
      acc11 = wmma_f32_k4(a1, b1, acc11);
    }
  };

  // ---------------- Encode: Y = z @ We1, K = 2048 in chunks of 32 -----------
#if defined(HAVE_ASYNC_LDS)
  // Double-buffered: async DMA fills buf^1 while WMMAs eat buf.
  auto stage_async = [&](int k0, int buf) {
    #pragma unroll
    for (int i = 0; i < 4; ++i) {
      int idx = tid + i * 256;
      int r = idx >> 3;            // 0..127
      int c = (idx & 7) << 2;      // 0,4,...,28
      size_t goff = (size_t)(blockRow + r) * DM + (size_t)(k0 + c);
      ASYNC_B128(z0 + goff, &lds[buf][0][r][c]);
      ASYNC_B128(z1 + goff, &lds[buf][1][r][c]);
    }
  };

  stage_async(0, 0);
  for (int k0 = 0; k0 < DM; k0 += 32) {
    int cur = (k0 >> 5) & 1;
    WAIT_ASYNC0();                 // my async fills of lds[cur] are done
    __syncthreads();               // everyone's fills visible; prev compute done
    if (k0 + 32 < DM) stage_async(k0 + 32, cur ^ 1);
    encode_step(cur, k0);
  }
#else
  for (int k0 = 0; k0 < DM; k0 += 32) {
    __syncthreads();
    #pragma unroll
    for (int i = 0; i < 4; ++i) {
      int idx = tid + i * 256;
      int r = idx >> 3;
      int c = (idx & 7) << 2;
      size_t goff = (size_t)(blockRow + r) * DM + (size_t)(k0 + c);
      float4 a = *(const float4*)(z0 + goff);
      float4 b = *(const float4*)(z1 + goff);
      *(float4*)&lds[0][0][r][c] = a;
      *(float4*)&lds[0][1][r][c] = b;
      if (k0 + 32 < DM) {
        __builtin_prefetch(z0 + goff + 32, 0, 1);
        __builtin_prefetch(z1 + goff + 32, 0, 1);
      }
    }
    __syncthreads();
    encode_step(0, k0);
  }
#endif

  // bias + GELU; park Y0/Y1 in LDS buf0 (wave-private rows; same-wave DS in-order)
  {
    float bA = be1[l16];
    float bB = be1[16 + l16];
    #pragma unroll
    for (int v = 0; v < 8; ++v) {
      int r = myRow + v + hi * 8;               // D-layout row
      lds[0][0][r][l16]      = gelu_exact(acc00[v] + bA);
      lds[0][0][r][16 + l16] = gelu_exact(acc01[v] + bB);
      lds[0][1][r][l16]      = gelu_exact(acc10[v] + bA);
      lds[0][1][r][16 + l16] = gelu_exact(acc11[v] + bB);
    }
  }

  // ---------------- H = Y @ We2 + be2 (K = 32, WMMA) ------------------------
  v8f h00 = zero8, h01 = zero8, h10 = zero8, h11 = zero8;
  #pragma unroll
  for (int ks = 0; ks < 8; ++ks) {
    int ko = ks * 4 + ko2;
    v2f a0 = *(const v2f*)&lds[0][0][myRow + l16][ko];
    v2f a1 = *(const v2f*)&lds[0][1][myRow + l16][ko];
    v2f b0, b1;
    b0.x = We2[ko * LAT + l16];
    b0.y = We2[(ko + 1) * LAT + l16];
    b1.x = We2[ko * LAT + 16 + l16];
    b1.y = We2[(ko + 1) * LAT + 16 + l16];
    h00 = wmma_f32_k4(a0, b0, h00);
    h01 = wmma_f32_k4(a0, b1, h01);
    h10 = wmma_f32_k4(a1, b0, h10);
    h11 = wmma_f32_k4(a1, b1, h11);
  }
  {
    float bA = be2[l16], bB = be2[16 + l16];
    #pragma unroll
    for (int v = 0; v < 8; ++v) {
      h00[v] += bA; h01[v] += bB; h10[v] += bA; h11[v] += bB;
    }
  }

  // ---------------- Batched slerp (per-row over 32 latent dims) -------------
  const float t = tp[0];
  float hm0[8], hm1[8];
  #pragma unroll
  for (int v = 0; v < 8; ++v) {
    float s00 = h00[v] * h00[v] + h01[v] * h01[v];
    float s11 = h10[v] * h10[v] + h11[v] * h11[v];
    float s01 = h00[v] * h10[v] + h01[v] * h11[v];
    #pragma unroll
    for (int m = 1; m <= 8; m <<= 1) {     // reduce within each 16-lane half
      s00 += __shfl_xor(s00, m, 32);
      s11 += __shfl_xor(s11, m, 32);
      s01 += __shfl_xor(s01, m, 32);
    }
    float n0 = sqrtf(s00), n1 = sqrtf(s11);
    float scale = 0.5f * (n0 + n1);
    float i0 = 1.0f / (n0 + EPSF);
    float i1 = 1.0f / (n1 + EPSF);
    float dot = s01 * i0 * i1;
    dot = fminf(fmaxf(dot, -1.0f + EPSF), 1.0f - EPSF);
    float theta = acosf(dot);
    float st = sinf(theta);
    bool small = fabsf(theta) < EPSF;
    float c0 = small ? (1.0f - t) : sinf((1.0f - t) * theta) / st;
    float c1 = small ? t          : sinf(t * theta) / st;
    float w0 = i0 * c0 * scale;
    float w1 = i1 * c1 * scale;
    hm0[v] = h00[v] * w0 + h10[v] * w1;
    hm1[v] = h01[v] * w0 + h11[v] * w1;
  }

  // ---------------- G = gelu(Hm @ Wd1 + bd1) (K = 32, WMMA) -----------------
  #pragma unroll
  for (int v = 0; v < 8; ++v) {
    int r = myRow + v + hi * 8;
    lds[0][0][r][l16]      = hm0[v];
    lds[0][0][r][16 + l16] = hm1[v];
  }
  v2f ha[8];
  #pragma unroll
  for (int ks = 0; ks < 8; ++ks)
    ha[ks] = *(const v2f*)&lds[0][0][myRow + l16][ks * 4 + ko2];

  v8f g0 = zero8, g1 = zero8;
  #pragma unroll
  for (int ks = 0; ks < 8; ++ks) {
    int ko = ks * 4 + ko2;
    v2f b0, b1;
    b0.x = Wd1[ko * LAT + l16];
    b0.y = Wd1[(ko + 1) * LAT + l16];
    b1.x = Wd1[ko * LAT + 16 + l16];
    b1.y = Wd1[(ko + 1) * LAT + 16 + l16];
    g0 = wmma_f32_k4(ha[ks], b0, g0);
    g1 = wmma_f32_k4(ha[ks], b1, g1);
  }
  {
    float bA = bd1[l16], bB = bd1[16 + l16];
    #pragma unroll
    for (int v = 0; v < 8; ++v) {
      int r = myRow + v + hi * 8;
      lds[0][1][r][l16]      = gelu_exact(g0[v] + bA);
      lds[0][1][r][16 + l16] = gelu_exact(g1[v] + bB);
    }
  }
  v2f ga[8];                                  // A-frags of G, reused 128x
  #pragma unroll
  for (int ks = 0; ks < 8; ++ks)
    ga[ks] = *(const v2f*)&lds[0][1][myRow + l16][ks * 4 + ko2];

  // ---------------- Out = G @ Wd2 + bd2 (N = 2048, K = 32) ------------------
  for (int n0 = 0; n0 < DM; n0 += 16) {
    v8f o = zero8;
    #pragma unroll
    for (int ks = 0; ks < 8; ++ks) {
      int ko = ks * 4 + ko2;
      v2f b;
      b.x = Wd2[ko * DM + n0 + l16];
      b.y = Wd2[(ko + 1) * DM + n0 + l16];
      o = wmma_f32_k4(ga[ks], b, o);
    }
    float bias = bd2[n0 + l16];
    size_t base = (size_t)(blockRow + myRow + hi * 8) * DM + (size_t)(n0 + l16);
    #pragma unroll
    for (int v = 0; v < 8; ++v)
      out[base + (size_t)v * DM] = o[v] + bias;
  }
}

extern "C" void kernel_launch(void* const* d_in, const int* in_sizes, int n_in,
                              void* d_out, int out_size, void* d_ws, size_t ws_size,
                              hipStream_t stream) {
  const float* z0  = (const float*)d_in[0];
  const float* z1  = (const float*)d_in[1];
  const float* t   = (const float*)d_in[2];
  const float* We1 = (const float*)d_in[3];
  const float* be1 = (const float*)d_in[4];
  const float* We2 = (const float*)d_in[5];
  const float* be2 = (const float*)d_in[6];
  const float* Wd1 = (const float*)d_in[7];
  const float* bd1 = (const float*)d_in[8];
  const float* Wd2 = (const float*)d_in[9];
  const float* bd2 = (const float*)d_in[10];
  float* out = (float*)d_out;

  dim3 grid(32768 / ROWS);   // 256 blocks
  dim3 block(256);           // 8 wave32 per WGP-resident block
  hipLaunchKernelGGL(manifold_fused, grid, block, 0, stream,
                     z0, z1, t, We1, be1, We2, be2, Wd1, bd1, Wd2, bd2, out);
}